// LegendreConv_32796370272929
// MI455X (gfx1250) — compile-verified
//
#include <hip/hip_runtime.h>

typedef float v2f __attribute__((ext_vector_type(2)));
typedef float v8f __attribute__((ext_vector_type(8)));

namespace {
constexpr int kN = 4, kC = 16, kO = 32, kH = 64, kW = 64;
constexpr int kK = kC * 9;    // 144 unfolded patch length
constexpr int kL = kH * kW;   // 4096 output positions per (n, o)
}

// One wave32 computes one 16(o) x 16(l) output tile of the min-plus conv.
// For each k = (c,i,j): S_k[o,l] = w[o,k] + p[k,l] is formed as a rank-2
// f32 WMMA outer sum (exact in f32), then folded into a running min on VALU.
//
// Lane-masking trick: lanes 16-31 of the A fragment feed columns K=2/3 and of
// the B fragment rows K=2/3. We pin B row K=2 (hi b.x) and A col K=3 (hi a.y)
// to 0.0, so finite garbage loaded into hi-lane a.x / b.y is multiplied by
// zero inside the WMMA. All loads are therefore unpredicated (no exec masks).
__global__ __launch_bounds__(256) void tropical_conv3x3_wmma(
    const float* __restrict__ x,     // (N, C, H, W)
    const float* __restrict__ wgt,   // (O, K)
    float* __restrict__ out)         // (N, O, H, W)
{
  const int lane = threadIdx.x & 31;
  const int wave = threadIdx.x >> 5;
  const int tile = (blockIdx.x << 3) + wave;   // 0..2047

  const int ltile = tile & 255;        // 256 l-tiles of 16 columns
  const int otile = (tile >> 8) & 1;   // 2 o-tiles of 16 channels
  const int n     = tile >> 9;         // batch index

  const int h     = ltile >> 2;        // output row (4 l-tiles per image row)
  const int c0    = (ltile & 3) << 4;  // first output column of tile
  const int obase = otile << 4;

  const bool lo  = lane < 16;
  const int  l16 = lane & 15;

  // A fragment (16x4 f32): VGPR0 = col K=0 (lanes 0-15) / K=2 (lanes 16-31)
  //                        VGPR1 = col K=1 (lanes 0-15) / K=3 (lanes 16-31)
  v2f a; a.x = 0.0f; a.y = lo ? 1.0f : 0.0f;   // K=1 = ones, K=3 = 0
  // B fragment (4x16 f32): VGPR0 = row K=0 / K=2, VGPR1 = row K=1 / K=3
  v2f b; b.x = lo ? 1.0f : 0.0f; b.y = 0.0f;   // K=0 = ones, K=2 = 0
  v8f czero = {0.f, 0.f, 0.f, 0.f, 0.f, 0.f, 0.f, 0.f};

  v8f acc;
  #pragma unroll
  for (int v = 0; v < 8; ++v) acc[v] = __builtin_inff();

  // Per-lane clamped element offsets and additive-zero pad masks; these are
  // tile constants and live in registers across the whole c-loop.
  int   off[3][3];
  float msk[3][3];
  #pragma unroll
  for (int i = 0; i < 3; ++i) {
    const int  row  = h + i - 1;
    const bool rOK  = (unsigned)row < (unsigned)kH;
    const int  rowc = rOK ? row : 0;
    #pragma unroll
    for (int j = 0; j < 3; ++j) {
      const int  col  = c0 + l16 + j - 1;
      const bool cOK  = (unsigned)col < (unsigned)kW;
      const int  colc = cOK ? col : 0;
      off[i][j] = rowc * kW + colc;
      msk[i][j] = (rOK && cOK) ? 1.0f : 0.0f;
    }
  }

  const float* wrow = wgt + (obase + l16) * kK;      // valid for all 32 lanes
  const float* xb   = x + (size_t)n * kC * (size_t)kL;

  #pragma unroll 1
  for (int c = 0; c < kC; ++c) {
    const float* xc = xb + c * kL;
    const float* wk = wrow + c * 9;
    #pragma unroll
    for (int i = 0; i < 3; ++i) {
      #pragma unroll
      for (int j = 0; j < 3; ++j) {
        const float pv = xc[off[i][j]] * msk[i][j];  // padded patch value
        const float wv = wk[i * 3 + j];              // w[o=obase+l16, k]
        a.x = wv;
        b.y = pv;
        // D = w[:,k] (x) 1  +  1 (x) p[k,:]   -- exact f32 outer sum
        v8f s = __builtin_amdgcn_wmma_f32_16x16x4_f32(
            false, a, false, b, (short)0, czero, false, false);
        #pragma unroll
        for (int v = 0; v < 8; ++v) acc[v] = fminf(acc[v], s[v]);
      }
    }
  }

  // C/D layout: VGPR v holds row M=v (lanes 0-15) / M=v+8 (lanes 16-31),
  // column N = lane % 16.
  const int oOut = obase + (lo ? 0 : 8);
  const int lOut = h * kW + c0 + l16;
  float* op = out + (size_t)(n * kO + oOut) * (size_t)kL + lOut;
  #pragma unroll
  for (int v = 0; v < 8; ++v) op[(size_t)v * kL] = acc[v];
}

extern "C" void kernel_launch(void* const* d_in, const int* in_sizes, int n_in,
                              void* d_out, int out_size, void* d_ws, size_t ws_size,
                              hipStream_t stream) {
  (void)in_sizes; (void)n_in; (void)d_ws; (void)ws_size; (void)out_size;
  const float* x   = (const float*)d_in[0];
  const float* wgt = (const float*)d_in[1];
  float* out       = (float*)d_out;

  // 4 batches * 2 o-tiles * 256 l-tiles = 2048 waves, 8 waves per block.
  dim3 block(256);
  dim3 grid((kN * 2 * 256) / 8);
  hipLaunchKernelGGL(tropical_conv3x3_wmma, grid, block, 0, stream, x, wgt, out);
}